// MoEGate_70781061038167
// MI455X (gfx1250) — compile-verified
//
#include <hip/hip_runtime.h>
#include <hip/hip_bf16.h>

// MoE gating for MI455X (gfx1250, wave32).
// logits = x[16384,2560] @ W^T[2560,4]  -> softmax(4) -> top-2 -> aux loss.
// Memory bound: 168 MB streamed once => ~7.2 us at 23.3 TB/s. GEMV runs on
// V_WMMA_F32_16X16X4_F32 (16 tokens/wave, K stepped by 4), hidden_states
// streamed with non-temporal loads so the 40 KB weight stays in cache.

#define EMBED   2560
#define NEXP    4
#define TOPK    2
#define ALPHA_F 0.01f

typedef __attribute__((ext_vector_type(2))) float v2f;
typedef __attribute__((ext_vector_type(8))) float v8f;

__global__ void moe_gate_zero_kernel(float* __restrict__ psum, int* __restrict__ cnt) {
    int t = threadIdx.x;
    if (t < NEXP) { psum[t] = 0.0f; cnt[t] = 0; }
}

__global__ __launch_bounds__(256)
void moe_gate_main_kernel(const float* __restrict__ x,
                          const float* __restrict__ w,
                          int*   __restrict__ out_idx,   // [T,2] int32
                          float* __restrict__ out_w,     // [T,2] f32
                          int*   __restrict__ out_row,   // [T,2] int32
                          float* __restrict__ ws_psum,   // [4]
                          int*   __restrict__ ws_cnt,    // [4]
                          int T)
{
    __shared__ float logitsLDS[8][16][NEXP];
    __shared__ float blkPsum[NEXP];
    __shared__ int   blkCnt[NEXP];

    const int tid  = threadIdx.x;
    const int wave = tid >> 5;
    const int lane = tid & 31;

    if (tid < NEXP) { blkPsum[tid] = 0.0f; blkCnt[tid] = 0; }
    __syncthreads();

    // 16 tokens per wave, 8 waves per block -> 128 tokens per block.
    const int tokenBase = blockIdx.x * 128 + wave * 16;

    // A-fragment (16x4 f32): lanes 0-15 hold M=lane, K={k0,k0+1};
    // lanes 16-31 hold M=lane-16, K={k0+2,k0+3}.  (ISA 7.12.2 layout)
    const int m  = lane & 15;
    const int kb = (lane >> 4) << 1;
    const float* __restrict__ arow = x + (size_t)(tokenBase + m) * EMBED + kb;

    // B-fragment (4x16 f32): B[k][n] = W[n][k]. Lanes with n>=4 load a
    // duplicate expert row (n&3) -> garbage D columns 4..15 that we never
    // read. Keeps EXEC all-ones through every WMMA, no masking.
    const int n = lane & 15;
    const float* __restrict__ brow = w + (size_t)(n & (NEXP - 1)) * EMBED + kb;

    v8f c = {0.f, 0.f, 0.f, 0.f, 0.f, 0.f, 0.f, 0.f};

    for (int k0 = 0; k0 < EMBED; k0 += 16) {        // 160 iters x 4 WMMA
#pragma unroll
        for (int u = 0; u < 4; ++u) {
            const int k = k0 + u * 4;
            // Stream hidden_states non-temporally (read exactly once).
            v2f a = __builtin_nontemporal_load((const v2f*)(arow + k));
            v2f b = *(const v2f*)(brow + k);        // 40 KB, cache-resident
            // (neg_a, A, neg_b, B, c_mod, C, reuse_a, reuse_b)
            c = __builtin_amdgcn_wmma_f32_16x16x4_f32(
                    false, a, false, b, (short)0, c, false, false);
        }
    }

    // Accumulator layout: VGPR r -> lanes 0-15: (M=r, N=lane),
    //                               lanes 16-31: (M=r+8, N=lane-16).
#pragma unroll
    for (int r = 0; r < 8; ++r) {
        const int mm = r + ((lane >> 4) << 3);
        if (n < NEXP) logitsLDS[wave][mm][n] = c[r];
    }
    __syncthreads();

    if (lane < 16) {
        const int t = tokenBase + lane;
        float l0 = logitsLDS[wave][lane][0];
        float l1 = logitsLDS[wave][lane][1];
        float l2 = logitsLDS[wave][lane][2];
        float l3 = logitsLDS[wave][lane][3];

        float mx = fmaxf(fmaxf(l0, l1), fmaxf(l2, l3));
        float e0 = __expf(l0 - mx);
        float e1 = __expf(l1 - mx);
        float e2 = __expf(l2 - mx);
        float e3 = __expf(l3 - mx);
        float inv = 1.0f / (e0 + e1 + e2 + e3);
        float p[NEXP] = { e0 * inv, e1 * inv, e2 * inv, e3 * inv };

        // top-2, strict '>' keeps lowest index on ties (lax.top_k semantics)
        int i0 = 0; float v0 = p[0];
#pragma unroll
        for (int e = 1; e < NEXP; ++e) if (p[e] > v0) { v0 = p[e]; i0 = e; }
        int i1 = -1; float v1 = -1.0f;
#pragma unroll
        for (int e = 0; e < NEXP; ++e) if (e != i0 && p[e] > v1) { v1 = p[e]; i1 = e; }

        out_idx[t * 2 + 0] = i0;
        out_idx[t * 2 + 1] = i1;
        out_w  [t * 2 + 0] = v0;
        out_w  [t * 2 + 1] = v1;
        out_row[t * 2 + 0] = t;          // row_idx[i][j] = j*T + i
        out_row[t * 2 + 1] = T + t;

        atomicAdd(&blkPsum[0], p[0]);
        atomicAdd(&blkPsum[1], p[1]);
        atomicAdd(&blkPsum[2], p[2]);
        atomicAdd(&blkPsum[3], p[3]);
        atomicAdd(&blkCnt[i0], 1);
        atomicAdd(&blkCnt[i1], 1);
    }
    __syncthreads();

    if (tid < NEXP) {
        atomicAdd(&ws_psum[tid], blkPsum[tid]);
        atomicAdd(&ws_cnt[tid],  blkCnt[tid]);
    }
}

__global__ void moe_gate_finalize_kernel(const float* __restrict__ psum,
                                         const int* __restrict__ cnt,
                                         float* __restrict__ aux_out, int T)
{
    if (threadIdx.x == 0 && blockIdx.x == 0) {
        float acc = 0.0f;
        const float invT  = 1.0f / (float)T;
        const float invTK = 1.0f / (float)(T * TOPK);
#pragma unroll
        for (int e = 0; e < NEXP; ++e) {
            float Pi = psum[e] * invT;                 // mean prob
            float fi = ((float)cnt[e] * invTK) * (float)NEXP;
            acc += Pi * fi;
        }
        aux_out[0] = acc * ALPHA_F;
    }
}

extern "C" void kernel_launch(void* const* d_in, const int* in_sizes, int n_in,
                              void* d_out, int out_size, void* d_ws, size_t ws_size,
                              hipStream_t stream) {
    const float* x = (const float*)d_in[0];   // [4,4096,2560] f32
    const float* w = (const float*)d_in[1];   // [4,2560] f32
    const int T = in_sizes[0] / EMBED;        // 16384

    // d_out layout (tuple, flat): topk_idx[T*2] (i32 bits), topk_weight[T*2],
    // row_idx[T*2] (i32 bits), aux_loss[1]
    float* outf    = (float*)d_out;
    int*   out_idx = (int*)outf;
    float* out_w   = outf + (size_t)T * 2;
    int*   out_row = (int*)(outf + (size_t)T * 4);
    float* aux     = outf + (size_t)T * 6;

    float* ws_psum = (float*)d_ws;
    int*   ws_cnt  = (int*)((char*)d_ws + NEXP * sizeof(float));

    moe_gate_zero_kernel<<<1, 32, 0, stream>>>(ws_psum, ws_cnt);
    moe_gate_main_kernel<<<T / 128, 256, 0, stream>>>(x, w, out_idx, out_w,
                                                      out_row, ws_psum, ws_cnt, T);
    moe_gate_finalize_kernel<<<1, 32, 0, stream>>>(ws_psum, ws_cnt, aux, T);
}